// MedianFilterLoss_86990267613346
// MI455X (gfx1250) — compile-verified
//
#include <hip/hip_runtime.h>
#include <hip/hip_bf16.h>
#include <math.h>

#define T_LEN   16384
#define THREADS 256
#define CHUNK   (T_LEN / THREADS)   // 64 timesteps per lane
#define BIGV    1e30f

static_assert(THREADS * CHUNK == T_LEN, "chunking must cover row");

// ---------------------------------------------------------------------------
// Transfer record for a contiguous chunk of one row's timeline.
// States 1..3 of the reference DP map to indices 0..2 (state 0 is always BIG).
// A[3][3]  : min-plus product of per-element transition matrices over the
//            chunk's LEADING run of ones only.
// s1, L    : sum of c1 / count over the leading ones.
// hasZero  : chunk contains at least one zero (out-state is then concrete).
// phase,p,dp : concrete automaton out-state (phase = min(run len,3),
//              p = pending sum of c1 of current run, dp = DP from run start).
// lossc    : loss contributions that do not depend on the incoming state
//            (c0 at zeros + commits of runs fully inside the chunk).
// ---------------------------------------------------------------------------
struct Rec {
  float A[9];
  float s1;
  float p;
  float d0, d1, d2;
  float lossc;
  int   L;
  int   hasZero;
  int   phase;
};

__device__ __forceinline__ float min3f(float a, float b, float c) {
  return fminf(a, fminf(b, c));
}

__device__ __forceinline__ float commitOf(int phase, float p,
                                          float d0, float d1, float d2) {
  // run length 0 -> 0 ; 1..2 -> forced (sum c1) ; >=3 -> min over DP states
  if (phase == 0) return 0.0f;
  if (phase < 3)  return p;
  return min3f(d0, d1, d2);
}

// C = A2 (applied second) o A1 (applied first), min-plus
__device__ __forceinline__ void matmulMP(float* __restrict__ C,
                                         const float* __restrict__ A2,
                                         const float* __restrict__ A1) {
#pragma unroll
  for (int i = 0; i < 3; ++i)
#pragma unroll
    for (int j = 0; j < 3; ++j) {
      float v = min3f(A2[i*3+0] + A1[0*3+j],
                      A2[i*3+1] + A1[1*3+j],
                      A2[i*3+2] + A1[2*3+j]);
      C[i*3+j] = fminf(v, BIGV);
    }
}

__device__ __forceinline__ void matvecMP(float* __restrict__ o,
                                         const float* __restrict__ A,
                                         float v0, float v1, float v2) {
#pragma unroll
  for (int i = 0; i < 3; ++i) {
    float v = min3f(A[i*3+0] + v0, A[i*3+1] + v1, A[i*3+2] + v2);
    o[i] = fminf(v, BIGV);
  }
}

// Compose: a = earlier chunk, b = later chunk. Exact and associative.
__device__ __forceinline__ Rec compose(const Rec& a, const Rec& b) {
  Rec c;
  if (!a.hasZero) {                 // a is all ones: extend leading run into b
    c.hasZero = b.hasZero;
    c.L  = a.L + b.L;
    c.s1 = a.s1 + b.s1;
    matmulMP(c.A, b.A, a.A);
    c.lossc = a.lossc + b.lossc;    // a.lossc == 0 by construction
    c.phase = b.phase; c.p = b.p;
    c.d0 = b.d0; c.d1 = b.d1; c.d2 = b.d2;
    return c;
  }
  // a has a zero: its out-state is concrete and feeds b's leading run.
  c.hasZero = 1;
  c.L = a.L; c.s1 = a.s1;
#pragma unroll
  for (int i = 0; i < 9; ++i) c.A[i] = a.A[i];

  if (b.L == 0) {                   // b starts with a zero: commit a's run now
    float bnd = commitOf(a.phase, a.p, a.d0, a.d1, a.d2);
    c.phase = b.phase; c.p = b.p;
    c.d0 = b.d0; c.d1 = b.d1; c.d2 = b.d2;
    c.lossc = a.lossc + b.lossc + bnd;
    return c;
  }
  if (b.hasZero) {                  // crossing run ends inside b
    float bnd;
    if (a.phase + b.L <= 2) {       // total run length 1..2 -> forced ones
      bnd = a.p + b.s1;
    } else {                        // DP run: dp_end = b.A (x) dp_start
      float e0, e1, e2;
      if (a.phase == 0) { e0 = b.A[2]; e1 = b.A[5]; e2 = b.A[8]; } // reset col
      else {
        float o[3]; matvecMP(o, b.A, a.d0, a.d1, a.d2);
        e0 = o[0]; e1 = o[1]; e2 = o[2];
      }
      bnd = min3f(e0, e1, e2);
    }
    c.phase = b.phase; c.p = b.p;
    c.d0 = b.d0; c.d1 = b.d1; c.d2 = b.d2;
    c.lossc = a.lossc + b.lossc + bnd;
    return c;
  }
  // b is all ones: no commit, extend out-state through b
  int np = a.phase + b.L; c.phase = (np > 3) ? 3 : np;
  c.p = a.p + b.s1;
  if (a.phase == 0) { c.d0 = b.A[2]; c.d1 = b.A[5]; c.d2 = b.A[8]; }
  else {
    float o[3]; matvecMP(o, b.A, a.d0, a.d1, a.d2);
    c.d0 = o[0]; c.d1 = o[1]; c.d2 = o[2];
  }
  c.lossc = a.lossc + b.lossc;
  return c;
}

// ---------------------------------------------------------------------------
// Kernel 1: one block per row; each lane streams 64 contiguous timesteps once
// building its transfer record; LDS tree-reduce 256 records -> row loss.
// ---------------------------------------------------------------------------
__global__ void __launch_bounds__(THREADS)
median_rows(const float* __restrict__ X, const int* __restrict__ TG,
            float* __restrict__ rowLoss) {
  const int row = blockIdx.x;
  const int tid = threadIdx.x;
  const long base = (long)row * T_LEN + (long)tid * CHUNK;

  // min-plus identity
  float A[9] = {0.f, BIGV, BIGV,  BIGV, 0.f, BIGV,  BIGV, BIGV, 0.f};
  float s1 = 0.f, p = 0.f, d0 = BIGV, d1 = BIGV, d2 = BIGV, lossc = 0.f;
  int L = 0, hasZero = 0, phase = 0;

  const float4* xv = (const float4*)(X + base);
  const int4*   tv = (const int4*)(TG + base);

  for (int j = 0; j < CHUNK / 4; ++j) {
    float4 xq = xv[j];
    int4   tq = tv[j];
    float xs[4] = {xq.x, xq.y, xq.z, xq.w};
    int   ts[4] = {tq.x, tq.y, tq.z, tq.w};
#pragma unroll
    for (int e = 0; e < 4; ++e) {
      const float x  = xs[e];
      const int   t  = ts[e];
      const float ax = fabsf(x);
      const float c0 = log1pf(__expf(-ax)) + fmaxf(x, 0.f); // softplus(x)
      const float c1 = c0 - x;                              // softplus(-x)
      if (!hasZero) {
        if (t) {
          // A = T(c0,c1) o A   (update leading-run transfer)
          float n10 = fminf(c1 + A[3], BIGV);
          float n11 = fminf(c1 + A[4], BIGV);
          float n12 = fminf(c1 + A[5], BIGV);
          float n20 = fminf(c0 + A[6], BIGV);
          float n21 = fminf(c0 + A[7], BIGV);
          float n22 = fminf(c0 + A[8], BIGV);
          float n30 = fminf(c1 + fminf(A[0], A[6]), BIGV);
          float n31 = fminf(c1 + fminf(A[1], A[7]), BIGV);
          float n32 = fminf(c1 + fminf(A[2], A[8]), BIGV);
          A[0]=n10; A[1]=n11; A[2]=n12;
          A[3]=n20; A[4]=n21; A[5]=n22;
          A[6]=n30; A[7]=n31; A[8]=n32;
          ++L; s1 += c1;
        } else {
          hasZero = 1; lossc += c0; phase = 0; p = 0.f;
        }
      } else {
        if (t) {
          if (phase == 0) {          // fresh run: reset then one transition
            d0 = BIGV; d1 = c0; d2 = c1; p = c1; phase = 1;
          } else {
            float n1 = fminf(d1 + c1, BIGV);
            float n2 = fminf(d2 + c0, BIGV);
            float n3 = fminf(fminf(d0, d2) + c1, BIGV);
            d0 = n1; d1 = n2; d2 = n3;
            p += c1; phase = (phase < 3) ? phase + 1 : 3;
          }
        } else {                     // run (or nothing) ends; zero cost
          lossc += c0 + commitOf(phase, p, d0, d1, d2);
          phase = 0; p = 0.f;
        }
      }
    }
  }

  __shared__ Rec sR[THREADS];
  Rec r;
#pragma unroll
  for (int i = 0; i < 9; ++i) r.A[i] = A[i];
  r.s1 = s1; r.p = p; r.d0 = d0; r.d1 = d1; r.d2 = d2;
  r.lossc = lossc; r.L = L; r.hasZero = hasZero; r.phase = phase;
  sR[tid] = r;
  __syncthreads();

  for (int s = THREADS / 2; s > 0; s >>= 1) {
    if (tid < s) sR[tid] = compose(sR[tid], sR[tid + s]);
    __syncthreads();
  }

  if (tid == 0) {
    Rec R = sR[0];
    // feed the virtual initial state (phase 0): commit of the row's leading run
    float b0 = 0.f;
    if (R.L > 0) {
      if (R.hasZero && R.L <= 2) b0 = R.s1;                       // forced
      else b0 = min3f(R.A[2], R.A[5], R.A[8]);                    // DP from reset
    }
    // trailing commit at sequence end
    float fc = R.hasZero ? commitOf(R.phase, R.p, R.d0, R.d1, R.d2) : 0.f;
    rowLoss[row] = R.lossc + b0 + fc;
  }
}

// ---------------------------------------------------------------------------
// Kernel 2: deterministic WMMA ones-matmul reduction of per-row losses.
// D = A_chunk(16x4 f32) x ones(4x16) + C accumulates row-sums in f32.
// ---------------------------------------------------------------------------
typedef __attribute__((ext_vector_type(2))) float v2f;
typedef __attribute__((ext_vector_type(8))) float v8f;

__global__ void reduce_rows(const float* __restrict__ rows,
                            float* __restrict__ out, int nrows, float inv_n) {
#if defined(__gfx1250__) && __has_builtin(__builtin_amdgcn_wmma_f32_16x16x4_f32)
  const int l = threadIdx.x;            // single wave32, EXEC all ones
  v8f acc = {};
  v2f bones; bones[0] = 1.0f; bones[1] = 1.0f;
  const int chunks = nrows / 64;
  const int m = (l < 16) ? l : (l - 16);
  const int k = (l < 16) ? 0 : 2;       // A 16x4 layout: VGPR0={K0,K2}, VGPR1={K1,K3}
  for (int q = 0; q < chunks; ++q) {
    const int idx = q * 64 + m * 4 + k;
    v2f a; a[0] = rows[idx]; a[1] = rows[idx + 1];
    acc = __builtin_amdgcn_wmma_f32_16x16x4_f32(
        /*neg_a=*/false, a, /*neg_b=*/false, bones,
        /*c_mod=*/(short)0, acc, /*reuse_a=*/false, /*reuse_b=*/false);
  }
  // D[i][j] = rowsum_i (any j). Lane 0 holds M=0..7 (N=0), lane 16 holds M=8..15.
  float s = acc[0] + acc[1] + acc[2] + acc[3] + acc[4] + acc[5] + acc[6] + acc[7];
  float total = __shfl(s, 0, 32) + __shfl(s, 16, 32);
  // scalar tail (nrows % 64), fixed order for determinism
  if (l == 0) {
    for (int i = chunks * 64; i < nrows; ++i) total += rows[i];
    out[0] = total * inv_n;
  }
#else
  const int l = threadIdx.x;
  float s = 0.f;
  for (int i = l; i < nrows; i += 32) s += rows[i];
  for (int off = 16; off > 0; off >>= 1) s += __shfl_down(s, off, 32);
  if (l == 0) out[0] = s * inv_n;
#endif
}

// ---------------------------------------------------------------------------
extern "C" void kernel_launch(void* const* d_in, const int* in_sizes, int n_in,
                              void* d_out, int out_size, void* d_ws, size_t ws_size,
                              hipStream_t stream) {
  const float* x  = (const float*)d_in[0];
  const int*   tg = (const int*)d_in[1];
  float* out      = (float*)d_out;
  float* rowloss  = (float*)d_ws;

  const long total = (long)in_sizes[0];
  const int  B     = (int)(total / T_LEN);

  median_rows<<<dim3(B), dim3(THREADS), 0, stream>>>(x, tg, rowloss);

  const float inv_n = 1.0f / (float)total;
  reduce_rows<<<dim3(1), dim3(32), 0, stream>>>(rowloss, out, B, inv_n);
}